// DeepElNet_6408091205737
// MI455X (gfx1250) — compile-verified
//
#include <hip/hip_runtime.h>
#include <hip/hip_bf16.h>

typedef __attribute__((ext_vector_type(2))) float v2f;
typedef __attribute__((ext_vector_type(4))) float v4f;
typedef __attribute__((ext_vector_type(8))) float v8f;

#define D 128

// ---------------------------------------------------------------- zero -----
__global__ void rgcn_zero(float* __restrict__ p, size_t count) {
    size_t i = (size_t)blockIdx.x * blockDim.x + threadIdx.x;
    size_t stride = (size_t)gridDim.x * blockDim.x;
    for (; i < count; i += stride) p[i] = 0.0f;
}

// ------------------------------------------------------------- scatter -----
// One wave (32 lanes) per edge: lane loads float4 of the source feature row
// and atomically accumulates into msg[(rel*N + dst)*D + ...]. Lane 0 counts.
__global__ void rgcn_scatter(const float* __restrict__ x,
                             const int* __restrict__ src,
                             const int* __restrict__ dst,
                             const int* __restrict__ rel,
                             float* __restrict__ msg,
                             float* __restrict__ cnt,
                             int N, int E) {
    int wavesPerBlock = blockDim.x >> 5;
    int wave = threadIdx.x >> 5;
    int lane = threadIdx.x & 31;
    int e = blockIdx.x * wavesPerBlock + wave;
    if (e >= E) return;

    int s = src[e];
    int d = dst[e];
    int r = rel[e];

    const float* xs = x + (size_t)s * D + lane * 4;
    v4f v = *(const v4f*)xs;

    float* m = msg + ((size_t)r * N + d) * D + lane * 4;
    atomicAdd(m + 0, v.x);
    atomicAdd(m + 1, v.y);
    atomicAdd(m + 2, v.z);
    atomicAdd(m + 3, v.w);
    if (lane == 0) atomicAdd(cnt + (size_t)r * N + d, 1.0f);
}

// -------------------------------------------------------------- invcnt -----
__global__ void rgcn_invcnt(float* __restrict__ cnt, size_t count) {
    size_t i = (size_t)blockIdx.x * blockDim.x + threadIdx.x;
    size_t stride = (size_t)gridDim.x * blockDim.x;
    for (; i < count; i += stride) {
        float c = cnt[i];
        cnt[i] = 1.0f / fmaxf(c, 1.0f);
    }
}

// ----------------------------------------------------- WMMA fused GEMM -----
// out[n, :] = relu( sum_r (msg[r,n,:] * invcnt[r,n]) @ W[r] + xin[n,:] @ root + bias )
// One wave computes a 32x128 output block: two 16-row A fragments share every
// B fragment (16 WMMAs per 8 ds_load_b64), halving LDS bandwidth per FLOP.
// B (64KB weight matrix) staged in LDS in a paired-K layout (conflict-free b64).
__global__ __launch_bounds__(256)
void rgcn_wmma_gemm(const float* __restrict__ msg,
                    const float* __restrict__ invcnt,
                    const float* __restrict__ xin,
                    const float* __restrict__ W,     // R x D x D
                    const float* __restrict__ root,  // D x D
                    const float* __restrict__ bias,  // D
                    float* __restrict__ out,         // N x D
                    int N, int R) {
    // LDS layout: element (k,c) at Bs[(k>>1)*256 + 2c + (k&1)]  (64 KB)
    __shared__ float Bs[D * D];

    const int lane    = threadIdx.x & 31;
    const int wave    = threadIdx.x >> 5;
    const int pairBlk = blockIdx.x * (blockDim.x >> 5) + wave;

    const int n0 = pairBlk * 32;            // first 16-row block
    const int n1 = n0 + 16;                 // second 16-row block
    const bool act1 = (n0 < N);             // wave-uniform
    const bool act2 = (n1 < N);             // wave-uniform

    const int row  = lane & 15;          // A row held by this lane
    const int koff = (lane >> 4) << 1;   // K sub-offset: 0 (lanes 0-15) / 2 (lanes 16-31)
    const int col  = lane & 15;          // B/D column held by this lane

    v8f acc1[8], acc2[8];
#pragma unroll
    for (int t = 0; t < 8; ++t) { acc1[t] = (v8f)0.0f; acc2[t] = (v8f)0.0f; }

    for (int r = 0; r <= R; ++r) {
        const bool isRoot = (r == R);
        const float* B = isRoot ? root : (W + (size_t)r * D * D);

        // ---- stage B into LDS (paired-K layout), all 256 threads ----
        __syncthreads();                 // previous iteration's reads complete
        for (int p = threadIdx.x; p < (D * D) / 2; p += 256) {
            int k2 = p >> 7;             // pair index: covers k = 2*k2, 2*k2+1
            int c  = p & (D - 1);
            v2f w;
            w.x = B[(size_t)(2 * k2) * D + c];
            w.y = B[(size_t)(2 * k2 + 1) * D + c];
            *(v2f*)&Bs[k2 * 256 + c * 2] = w;   // ds_store_b64, conflict-free
        }
        __syncthreads();

        if (act1) {
            const float* A0 = isRoot ? (xin + (size_t)n0 * D)
                                     : (msg + ((size_t)r * N + n0) * D);
            const float s1 = isRoot ? 1.0f : invcnt[(size_t)r * N + n0 + row];
            const float* A1row = A0 + (size_t)row * D;

            if (act2) {
                const float* A2row = A1row + (size_t)16 * D;
                const float s2 = isRoot ? 1.0f : invcnt[(size_t)r * N + n1 + row];
                for (int k = 0; k < D; k += 4) {
                    const int ka = k + koff;                 // always even
                    v2f av1 = *(const v2f*)(A1row + ka);
                    v2f av2 = *(const v2f*)(A2row + ka);
                    av1.x *= s1; av1.y *= s1;
                    av2.x *= s2; av2.y *= s2;
                    const float* bp = &Bs[(ka >> 1) * 256 + col * 2];
#pragma unroll
                    for (int t = 0; t < 8; ++t) {
                        v2f bv = *(const v2f*)(bp + t * 32);  // ds_load_b64
                        acc1[t] = __builtin_amdgcn_wmma_f32_16x16x4_f32(
                            false, av1, false, bv, (short)0, acc1[t], false, false);
                        acc2[t] = __builtin_amdgcn_wmma_f32_16x16x4_f32(
                            false, av2, false, bv, (short)0, acc2[t], false, false);
                    }
                }
            } else {
                for (int k = 0; k < D; k += 4) {
                    const int ka = k + koff;
                    v2f av1 = *(const v2f*)(A1row + ka);
                    av1.x *= s1; av1.y *= s1;
                    const float* bp = &Bs[(ka >> 1) * 256 + col * 2];
#pragma unroll
                    for (int t = 0; t < 8; ++t) {
                        v2f bv = *(const v2f*)(bp + t * 32);
                        acc1[t] = __builtin_amdgcn_wmma_f32_16x16x4_f32(
                            false, av1, false, bv, (short)0, acc1[t], false, false);
                    }
                }
            }
        }
    }

    // Epilogue: C/D layout — VGPR v, lanes 0-15 => M=v, lanes 16-31 => M=v+8.
    const int mbase = (lane >> 4) * 8;
    if (act1) {
#pragma unroll
        for (int t = 0; t < 8; ++t) {
            float b = bias[t * 16 + col];
#pragma unroll
            for (int v = 0; v < 8; ++v) {
                float val = fmaxf(acc1[t][v] + b, 0.0f);
                out[(size_t)(n0 + mbase + v) * D + t * 16 + col] = val;
            }
        }
    }
    if (act2) {
#pragma unroll
        for (int t = 0; t < 8; ++t) {
            float b = bias[t * 16 + col];
#pragma unroll
            for (int v = 0; v < 8; ++v) {
                float val = fmaxf(acc2[t][v] + b, 0.0f);
                out[(size_t)(n1 + mbase + v) * D + t * 16 + col] = val;
            }
        }
    }
}

// ------------------------------------------------------------- launch ------
extern "C" void kernel_launch(void* const* d_in, const int* in_sizes, int n_in,
                              void* d_out, int out_size, void* d_ws, size_t ws_size,
                              hipStream_t stream) {
    const float* x     = (const float*)d_in[0];
    const int*   eidx  = (const int*)d_in[1];
    const int*   etype = (const int*)d_in[2];
    const float* W1    = (const float*)d_in[3];
    const float* root1 = (const float*)d_in[4];
    const float* b1    = (const float*)d_in[5];
    const float* W2    = (const float*)d_in[6];
    const float* root2 = (const float*)d_in[7];
    const float* b2    = (const float*)d_in[8];
    float* out = (float*)d_out;

    const int N = in_sizes[0] / D;
    const int E = in_sizes[2];
    const int R = in_sizes[3] / (D * D);

    const int* src = eidx;
    const int* dst = eidx + E;

    float* msg = (float*)d_ws;                   // R*N*D
    float* cnt = msg + (size_t)R * N * D;        // R*N
    float* h   = cnt + (size_t)R * N;            // N*D

    const size_t accumCount = (size_t)R * N * D + (size_t)R * N;

    dim3 zgrid(2048), zblk(256);
    dim3 sgrid((E + 7) / 8), sblk(256);          // 8 waves/block, 1 edge/wave
    dim3 igrid(1024), iblk(256);
    int pairBlks = (N + 31) / 32;                // 32 rows per wave
    dim3 ggrid((pairBlks + 7) / 8), gblk(256);   // 8 waves/block

    // ---- Layer 1: x -> h ----
    rgcn_zero<<<zgrid, zblk, 0, stream>>>(msg, accumCount);
    rgcn_scatter<<<sgrid, sblk, 0, stream>>>(x, src, dst, etype, msg, cnt, N, E);
    rgcn_invcnt<<<igrid, iblk, 0, stream>>>(cnt, (size_t)R * N);
    rgcn_wmma_gemm<<<ggrid, gblk, 0, stream>>>(msg, cnt, x, W1, root1, b1, h, N, R);

    // ---- Layer 2: h -> out ----
    rgcn_zero<<<zgrid, zblk, 0, stream>>>(msg, accumCount);
    rgcn_scatter<<<sgrid, sblk, 0, stream>>>(h, src, dst, etype, msg, cnt, N, E);
    rgcn_invcnt<<<igrid, iblk, 0, stream>>>(cnt, (size_t)R * N);
    rgcn_wmma_gemm<<<ggrid, gblk, 0, stream>>>(msg, cnt, h, W2, root2, b2, out, N, R);
}